// ContextDrivingForce_5488968204925
// MI455X (gfx1250) — compile-verified
//
#include <hip/hip_runtime.h>
#include <math.h>

// ContextDrivingForce fused pipeline for MI455X (gfx1250), fp32 end-to-end.
// Memory-bound (AI ~650 FLOP/B) => keep fp32 precision, V_WMMA_F32_16X16X4_F32.
// B tiles staged into LDS by the Tensor Data Mover (TENSOR_LOAD_TO_LDS).

#define NTOK 16384
#define DIM  896
#define MT   64    // rows per workgroup
#define NTB  128   // cols per block tile
#define KC   32    // K chunk
#define KP   (KC + 2)  // padded LDS row stride (34 dwords): b64-aligned, conflict-free

typedef __attribute__((ext_vector_type(2))) float v2f;
typedef __attribute__((ext_vector_type(8))) float v8f;
typedef __attribute__((ext_vector_type(4))) unsigned int v4u;
typedef __attribute__((ext_vector_type(8))) unsigned int v8u;

__device__ __forceinline__ v8f wmma_f32(v2f a, v2f b, v8f c) {
  // (neg_a, A, neg_b, B, c_mod, C, reuse_a, reuse_b)
  return __builtin_amdgcn_wmma_f32_16x16x4_f32(false, a, false, b, (short)0, c,
                                               false, false);
}

__device__ __forceinline__ float sigmoidf_(float x) {
  return 1.0f / (1.0f + expf(-x));
}

// ---------------------------------------------------------------------------
// TDM: async 2D block copy global->LDS (CDNA5 ISA-8).
// Copies n_rows x KC f32 elements, global row stride = row_stride elems,
// into LDS rows padded to KP dwords (pad_interval=32 dwords, pad_amount=2).
// Descriptor groups 2/3 omitted (2D tensor). Issue from one wave only;
// fence with s_wait_tensorcnt before the workgroup barrier.
// ---------------------------------------------------------------------------
__device__ __forceinline__ void tdm_load_2d(unsigned int lds_byte_off,
                                            const float* gptr,
                                            unsigned int n_rows,
                                            unsigned long long row_stride) {
  unsigned long long ga = (unsigned long long)(uintptr_t)gptr;
  v4u g0;
  g0[0] = 1u;                                           // count=1, user D#
  g0[1] = lds_byte_off;                                 // lds_addr
  g0[2] = (unsigned int)(ga & 0xFFFFFFFFu);             // global_addr[31:0]
  g0[3] = (unsigned int)((ga >> 32) & 0x01FFFFFFu)      // global_addr[56:32]
          | (2u << 30);                                 // type=2 (image)
  v8u g1;
  g1[0] = (2u << 16)      // data_size = 4B
        | (1u << 20)      // pad_enable
        | (4u << 22)      // pad_interval: 32 dwords
        | (1u << 25);     // pad_amount: 2 dwords  -> LDS row stride 34 dwords
  g1[1] = ((unsigned int)KC & 0xFFFFu) << 16;           // tensor_dim0 lo (=KC)
  g1[2] = ((n_rows & 0xFFFFu) << 16);                   // dim0 hi=0 | tensor_dim1 lo
  g1[3] = ((n_rows >> 16) & 0xFFFFu)                    // tensor_dim1 hi
        | (((unsigned int)KC & 0xFFFFu) << 16);         // tile_dim0 = KC
  g1[4] = (n_rows & 0xFFFFu);                           // tile_dim1 = n_rows, tile_dim2=0
  g1[5] = (unsigned int)(row_stride & 0xFFFFFFFFu);     // tensor_dim0_stride lo
  g1[6] = (unsigned int)((row_stride >> 32) & 0xFFFFu); // stride hi | dim1_stride lo=0
  g1[7] = 0u;
  asm volatile("tensor_load_to_lds %0, %1" :: "s"(g0), "s"(g1) : "memory");
}

// ---------------------------------------------------------------------------
// Kernel A: logit[m] = sum_n h[m,n]*(u@Wa^T + ba)[m,n];
//           u_att = sigmoid(logit/sqrt(D)) * u     -> d_out
// One workgroup owns 64 full rows so the row reduction stays on-chip;
// u_proj is never materialized.
// ---------------------------------------------------------------------------
__global__ __launch_bounds__(256) void kA(const float* __restrict__ h,
                                          const float* __restrict__ u,
                                          const float* __restrict__ Wa,
                                          const float* __restrict__ ba,
                                          float* __restrict__ uatt) {
  __shared__ __align__(16) float As[MT][KP];
  __shared__ __align__(16) float Bs[NTB][KP];   // n-major: Bs[n][k] = Wa[n0+n][k0+k]
  __shared__ float logit[MT];
  __shared__ float alpha[MT];

  const int tid  = threadIdx.x;
  const int lane = tid & 31;
  const int wv   = tid >> 5;
  const int half = lane >> 4;
  const int l    = lane & 15;
  const int miw  = (wv & 3) * 16;   // 4 x 16 = 64 rows
  const int niw  = (wv >> 2) * 64;  // 2 x 64 = 128 cols
  const size_t row0 = (size_t)blockIdx.x * MT;
  const unsigned int bs_off = (unsigned int)(uintptr_t)(&Bs[0][0]);

  if (tid < MT) logit[tid] = 0.0f;
  __syncthreads();

  for (int nb = 0; nb < DIM / NTB; ++nb) {
    const int n0 = nb * NTB;
    v8f acc[4];
    for (int t = 0; t < 4; ++t) acc[t] = {};

    for (int k0 = 0; k0 < DIM; k0 += KC) {
      __syncthreads();
      if (wv == 0)  // TDM: Wa tile [128 rows x 32 f32], row stride DIM
        tdm_load_2d(bs_off, Wa + (size_t)n0 * DIM + k0, NTB, DIM);
      // A tile: rows of u_t (coalesced 128B per 32 threads)
      for (int idx = tid; idx < MT * KC; idx += 256) {
        int m = idx >> 5, k = idx & 31;
        As[m][k] = u[(row0 + m) * DIM + (k0 + k)];
      }
      if (wv == 0) __builtin_amdgcn_s_wait_tensorcnt(0);
      __syncthreads();

      for (int kk = 0; kk < KC; kk += 4) {
        // A 16x4 f32: VGPR0={K0|K2}, VGPR1={K1|K3} across lane halves.
        v2f a = *(const v2f*)&As[miw + l][kk + 2 * half];
#pragma unroll
        for (int t = 0; t < 4; ++t) {
          v2f b = *(const v2f*)&Bs[niw + t * 16 + l][kk + 2 * half];
          acc[t] = wmma_f32(a, b, acc[t]);
        }
      }
    }

    // Fold this 128-col block into the per-row attention logit.
#pragma unroll
    for (int t = 0; t < 4; ++t) {
#pragma unroll
      for (int r = 0; r < 8; ++r) {
        int m   = miw + r + 8 * half;               // C/D layout row
        int col = n0 + niw + t * 16 + l;
        float p = (acc[t][r] + ba[col]) * h[(row0 + m) * DIM + col];
        p += __shfl_xor(p, 1, 16);
        p += __shfl_xor(p, 2, 16);
        p += __shfl_xor(p, 4, 16);
        p += __shfl_xor(p, 8, 16);
        if (l == 0) atomicAdd(&logit[m], p);        // ds_add_f32
      }
    }
  }
  __syncthreads();
  if (tid < MT) {
    alpha[tid] = sigmoidf_(logit[tid] * 0.033407655f);  // 1/sqrt(896)
  }
  __syncthreads();
  for (int idx = tid; idx < MT * DIM; idx += 256) {
    int m = idx / DIM, c = idx % DIM;
    uatt[(row0 + m) * DIM + c] = alpha[m] * u[(row0 + m) * DIM + c];
  }
}

// ---------------------------------------------------------------------------
// Kernel B/C: fused GEMM  y = A@W^T + bias,  A gathered on the fly.
// MODE 0: A=[h | x2],          out = sigmoid(y) * x2          (u_gate)
// MODE 1: A=[h | x2 | h*x2],   out = exact-erf GELU(y)        (g_t)
// out buffer != x2 buffer => no hazards; 2D grid.
// ---------------------------------------------------------------------------
template <int MODE, int KTOT>
__global__ __launch_bounds__(256) void kBC(const float* __restrict__ h,
                                           const float* __restrict__ x2,
                                           const float* __restrict__ W,
                                           const float* __restrict__ bias,
                                           float* __restrict__ out) {
  __shared__ __align__(16) float As[MT][KP];
  __shared__ __align__(16) float Bs[NTB][KP];

  const int tid  = threadIdx.x;
  const int lane = tid & 31;
  const int wv   = tid >> 5;
  const int half = lane >> 4;
  const int l    = lane & 15;
  const int miw  = (wv & 3) * 16;
  const int niw  = (wv >> 2) * 64;
  const size_t row0 = (size_t)blockIdx.x * MT;
  const int n0 = blockIdx.y * NTB;
  const unsigned int bs_off = (unsigned int)(uintptr_t)(&Bs[0][0]);

  v8f acc[4];
  for (int t = 0; t < 4; ++t) acc[t] = {};

  for (int k0 = 0; k0 < KTOT; k0 += KC) {
    __syncthreads();
    if (wv == 0)  // TDM: W tile [128 rows x 32 f32], row stride KTOT
      tdm_load_2d(bs_off, W + (size_t)n0 * KTOT + k0, NTB, KTOT);
    for (int idx = tid; idx < MT * KC; idx += 256) {
      int m = idx >> 5, k = idx & 31;
      int gk = k0 + k;
      size_t rb = (row0 + m) * DIM;
      float v;
      if (MODE == 0) {
        v = (gk < DIM) ? h[rb + gk] : x2[rb + (gk - DIM)];
      } else {
        if (gk < DIM)            v = h[rb + gk];
        else if (gk < 2 * DIM)   v = x2[rb + (gk - DIM)];
        else { int k2 = gk - 2 * DIM; v = h[rb + k2] * x2[rb + k2]; }
      }
      As[m][k] = v;
    }
    if (wv == 0) __builtin_amdgcn_s_wait_tensorcnt(0);
    __syncthreads();

    for (int kk = 0; kk < KC; kk += 4) {
      v2f a = *(const v2f*)&As[miw + l][kk + 2 * half];
#pragma unroll
      for (int t = 0; t < 4; ++t) {
        v2f b = *(const v2f*)&Bs[niw + t * 16 + l][kk + 2 * half];
        acc[t] = wmma_f32(a, b, acc[t]);
      }
    }
  }

#pragma unroll
  for (int t = 0; t < 4; ++t) {
#pragma unroll
    for (int r = 0; r < 8; ++r) {
      int m   = miw + r + 8 * half;
      int col = n0 + niw + t * 16 + l;
      float y = acc[t][r] + bias[col];
      float o;
      if (MODE == 0) {
        o = sigmoidf_(y) * x2[(row0 + m) * DIM + col];
      } else {
        o = 0.5f * y * (1.0f + erff(y * 0.70710678118f));  // exact GELU
      }
      out[(row0 + m) * DIM + col] = o;
    }
  }
}

extern "C" void kernel_launch(void* const* d_in, const int* in_sizes, int n_in,
                              void* d_out, int out_size, void* d_ws, size_t ws_size,
                              hipStream_t stream) {
  (void)in_sizes; (void)n_in; (void)out_size; (void)ws_size;
  const float* h  = (const float*)d_in[0];  // h_t  [N,D]
  const float* u  = (const float*)d_in[1];  // u_t  [N,D]
  // d_in[2]=token_idx (unused), d_in[3]=u_all (unused)
  const float* Wa = (const float*)d_in[4];  // [D, D]
  const float* ba = (const float*)d_in[5];
  const float* Wg = (const float*)d_in[6];  // [D, 2D]
  const float* bg = (const float*)d_in[7];
  const float* Wf = (const float*)d_in[8];  // [D, 3D]
  const float* bf = (const float*)d_in[9];

  float* outp = (float*)d_out;          // stage1: u_att, final: g_t
  float* ug   = (float*)d_ws;           // u_gate scratch: N*D floats (58.7 MB)

  kA<<<NTOK / MT, 256, 0, stream>>>(h, u, Wa, ba, outp);                  // u_att -> d_out
  kBC<0, 2 * DIM><<<dim3(NTOK / MT, DIM / NTB), 256, 0, stream>>>(h, outp, Wg, bg, ug);   // u_gate -> ws
  kBC<1, 3 * DIM><<<dim3(NTOK / MT, DIM / NTB), 256, 0, stream>>>(h, ug, Wf, bf, outp);   // g_t -> d_out
}